// GCNNetwork_15487652069901
// MI455X (gfx1250) — compile-verified
//
#include <hip/hip_runtime.h>
#include <hip/hip_bf16.h>
#include <math.h>

typedef float v2f __attribute__((ext_vector_type(2)));
typedef float v8f __attribute__((ext_vector_type(8)));

#if defined(__has_builtin)
#if __has_builtin(__builtin_amdgcn_wmma_f32_16x16x4_f32)
#define HAS_WMMA_F32_4 1
#endif
#endif
#ifndef HAS_WMMA_F32_4
#define HAS_WMMA_F32_4 0
#endif

#define FEAT 128
#define BN_EPS 1e-5f

// ---------------------------------------------------------------------------
// deg[col[e]] += w[e]   (fp32 relaxed agent-scope atomic -> global_atomic_add_f32)
// ---------------------------------------------------------------------------
__global__ __launch_bounds__(256) void degree_kernel(const int* __restrict__ col,
                                                     const float* __restrict__ w,
                                                     float* __restrict__ deg, int E) {
    int e = blockIdx.x * 256 + threadIdx.x;
    if (e < E) {
        __hip_atomic_fetch_add(&deg[col[e]], w[e], __ATOMIC_RELAXED,
                               __HIP_MEMORY_SCOPE_AGENT);
    }
}

// dis[i] = rsqrt(deg[i] + 1)   (self-loop weight 1.0 makes deg strictly > 0)
__global__ __launch_bounds__(256) void dis_kernel(float* __restrict__ deg_dis, int n) {
    int i = blockIdx.x * 256 + threadIdx.x;
    if (i < n) deg_dis[i] = rsqrtf(deg_dis[i] + 1.0f);
}

// ---------------------------------------------------------------------------
// Y[N,128] = X[N,128] @ W[128,128]  — fp32 WMMA 16x16x4.
// Block = 256 threads = 8 waves. Block handles 16 rows; wave w owns N-tile w.
// A tile staged in LDS (stride 132 to kill bank conflicts); B frags from global
// (W is 64KB, resident in WGP$/L2).
// Fragment layouts (wave32):
//   A 16x4:  lane = 16*kh + m ; v0 = A[m][k0+2*kh], v1 = A[m][k0+2*kh+1]
//   B 4x16:  lane = 16*kh + n ; v0 = W[k0+2*kh][n], v1 = W[k0+2*kh+1][n]
//   C 16x16: lanes 0-15: M = vgpr, N = lane ; lanes 16-31: M = vgpr+8, N = lane-16
// ---------------------------------------------------------------------------
__global__ __launch_bounds__(256) void gemm_n128_kernel(const float* __restrict__ X,
                                                        const float* __restrict__ W,
                                                        float* __restrict__ Y,
                                                        int nrows) {
    __shared__ float lA[16 * 132];

    const int tid = threadIdx.x;
    const int rowBase = blockIdx.x * 16;

    // Stage the 16x128 A tile, coalesced float4 loads (2 per thread).
#pragma unroll
    for (int i = 0; i < 2; ++i) {
        int idx = tid + i * 256;       // float4 index within the tile (0..511)
        int r = idx >> 5;              // 32 float4 per row
        int c4 = (idx & 31) << 2;
        float4 v = make_float4(0.f, 0.f, 0.f, 0.f);
        if (rowBase + r < nrows)
            v = *(const float4*)(X + (size_t)(rowBase + r) * FEAT + c4);
        float* dst = &lA[r * 132 + c4];
        dst[0] = v.x; dst[1] = v.y; dst[2] = v.z; dst[3] = v.w;
    }
    __syncthreads();

    const int wave = tid >> 5;     // n-tile index, 0..7
    const int lane = tid & 31;
    const int lid  = lane & 15;
    const int half = lane >> 4;
    const int n0   = wave * 16;

#if HAS_WMMA_F32_4
    v8f acc = {0.f, 0.f, 0.f, 0.f, 0.f, 0.f, 0.f, 0.f};
    const float* arow = &lA[lid * 132];
#pragma unroll
    for (int k0 = 0; k0 < FEAT; k0 += 4) {
        const int ka = k0 + 2 * half;
        v2f a, b;
        a.x = arow[ka];
        a.y = arow[ka + 1];
        b.x = W[(size_t)ka * FEAT + n0 + lid];
        b.y = W[(size_t)(ka + 1) * FEAT + n0 + lid];
        acc = __builtin_amdgcn_wmma_f32_16x16x4_f32(
            /*neg_a=*/false, a, /*neg_b=*/false, b,
            /*c_mod=*/(short)0, acc, /*reuse_a=*/false, /*reuse_b=*/false);
    }
    float* yb = Y + (size_t)rowBase * FEAT + n0 + lid;
#pragma unroll
    for (int v = 0; v < 8; ++v) {
        int m = v + 8 * half;
        if (rowBase + m < nrows) yb[(size_t)m * FEAT] = acc[v];
    }
#else
    // Scalar fallback (only if the fp32 WMMA builtin is unavailable).
    float accs[8] = {0.f, 0.f, 0.f, 0.f, 0.f, 0.f, 0.f, 0.f};
    for (int k = 0; k < FEAT; ++k) {
        float bv = W[(size_t)k * FEAT + n0 + lid];
#pragma unroll
        for (int v = 0; v < 8; ++v)
            accs[v] = fmaf(lA[(v + 8 * half) * 132 + k], bv, accs[v]);
    }
    float* yb = Y + (size_t)rowBase * FEAT + n0 + lid;
#pragma unroll
    for (int v = 0; v < 8; ++v) {
        int m = v + 8 * half;
        if (rowBase + m < nrows) yb[(size_t)m * FEAT] = accs[v];
    }
#endif
}

// ---------------------------------------------------------------------------
// agg[col[e]] += dis[row]*w*dis[col] * h[row]   (one wave per edge; float4/lane)
// ---------------------------------------------------------------------------
__global__ __launch_bounds__(256) void aggregate_kernel(const int* __restrict__ row,
                                                        const int* __restrict__ col,
                                                        const float* __restrict__ ew,
                                                        const float* __restrict__ dis,
                                                        const float* __restrict__ h,
                                                        float* __restrict__ agg, int E) {
    int gid = blockIdx.x * 256 + threadIdx.x;
    int e = gid >> 5;
    if (e >= E) return;
    int lane = gid & 31;
    int r = row[e];
    int c = col[e];
    float nrm = dis[r] * ew[e] * dis[c];
    const float4 hv = *(const float4*)(h + (size_t)r * FEAT + lane * 4);
    float* dst = agg + (size_t)c * FEAT + lane * 4;
    __hip_atomic_fetch_add(dst + 0, hv.x * nrm, __ATOMIC_RELAXED, __HIP_MEMORY_SCOPE_AGENT);
    __hip_atomic_fetch_add(dst + 1, hv.y * nrm, __ATOMIC_RELAXED, __HIP_MEMORY_SCOPE_AGENT);
    __hip_atomic_fetch_add(dst + 2, hv.z * nrm, __ATOMIC_RELAXED, __HIP_MEMORY_SCOPE_AGENT);
    __hip_atomic_fetch_add(dst + 3, hv.w * nrm, __ATOMIC_RELAXED, __HIP_MEMORY_SCOPE_AGENT);
}

// ---------------------------------------------------------------------------
// h1 = elu( BN( agg1 + dis[i]^2 * h + b0 ) )        (in-place on agg1)
// ---------------------------------------------------------------------------
__global__ __launch_bounds__(256) void epilogue1_kernel(float* __restrict__ agg,
                                                        const float* __restrict__ h,
                                                        const float* __restrict__ dis,
                                                        const float* __restrict__ b0,
                                                        const float* __restrict__ g,
                                                        const float* __restrict__ be,
                                                        const float* __restrict__ mu,
                                                        const float* __restrict__ var,
                                                        int n4) {
    int idx4 = blockIdx.x * 256 + threadIdx.x;
    if (idx4 >= n4) return;
    int i  = idx4 >> 5;          // node (32 float4 per row)
    int f  = (idx4 & 31) << 2;   // feature base
    float d = dis[i];
    float d2 = d * d;
    float4 a  = *(float4*)(agg + (size_t)idx4 * 4);
    float4 hv = *(const float4*)(h + (size_t)idx4 * 4);
    float out[4];
    float av[4] = {a.x, a.y, a.z, a.w};
    float hh[4] = {hv.x, hv.y, hv.z, hv.w};
#pragma unroll
    for (int j = 0; j < 4; ++j) {
        int ff = f + j;
        float v = av[j] + d2 * hh[j] + b0[ff];
        v = (v - mu[ff]) * rsqrtf(var[ff] + BN_EPS) * g[ff] + be[ff];
        out[j] = (v > 0.0f) ? v : expm1f(v);
    }
    *(float4*)(agg + (size_t)idx4 * 4) = make_float4(out[0], out[1], out[2], out[3]);
}

// ---------------------------------------------------------------------------
// out = ( emb + h1 + (agg2 + dis[i]^2 * h2pre + b1) ) / 3
// ---------------------------------------------------------------------------
__global__ __launch_bounds__(256) void final_kernel(const float* __restrict__ emb,
                                                    const float* __restrict__ h1,
                                                    const float* __restrict__ agg2,
                                                    const float* __restrict__ h2,
                                                    const float* __restrict__ dis,
                                                    const float* __restrict__ b1,
                                                    float* __restrict__ out, int n4) {
    int idx4 = blockIdx.x * 256 + threadIdx.x;
    if (idx4 >= n4) return;
    int i = idx4 >> 5;
    int f = (idx4 & 31) << 2;
    float d = dis[i];
    float d2 = d * d;
    float4 e  = *(const float4*)(emb  + (size_t)idx4 * 4);
    float4 o1 = *(const float4*)(h1   + (size_t)idx4 * 4);
    float4 a2 = *(const float4*)(agg2 + (size_t)idx4 * 4);
    float4 p2 = *(const float4*)(h2   + (size_t)idx4 * 4);
    const float inv3 = 1.0f / 3.0f;
    float4 r;
    r.x = (e.x + o1.x + a2.x + d2 * p2.x + b1[f + 0]) * inv3;
    r.y = (e.y + o1.y + a2.y + d2 * p2.y + b1[f + 1]) * inv3;
    r.z = (e.z + o1.z + a2.z + d2 * p2.z + b1[f + 2]) * inv3;
    r.w = (e.w + o1.w + a2.w + d2 * p2.w + b1[f + 3]) * inv3;
    *(float4*)(out + (size_t)idx4 * 4) = r;
}

// ---------------------------------------------------------------------------
extern "C" void kernel_launch(void* const* d_in, const int* in_sizes, int n_in,
                              void* d_out, int out_size, void* d_ws, size_t ws_size,
                              hipStream_t stream) {
    const float* emb = (const float*)d_in[0];
    const int*   ei  = (const int*)d_in[1];
    const float* ew  = (const float*)d_in[2];
    const float* W0  = (const float*)d_in[3];
    const float* b0  = (const float*)d_in[4];
    const float* W1  = (const float*)d_in[5];
    const float* b1  = (const float*)d_in[6];
    const float* bng = (const float*)d_in[7];
    const float* bnb = (const float*)d_in[8];
    const float* bnm = (const float*)d_in[9];
    const float* bnv = (const float*)d_in[10];
    float* out = (float*)d_out;

    const int N  = in_sizes[0] / FEAT;
    const int E  = in_sizes[2];
    const size_t ND = (size_t)N * FEAT;

    const int* row = ei;       // edge_index[0]
    const int* col = ei + E;   // edge_index[1]

    // Workspace layout (floats): [dis | bufA(h / h2pre) | bufB(agg1 -> h1) | bufC(agg2)]
    float* ws   = (float*)d_ws;
    size_t disPad = ((size_t)N + 1023) & ~(size_t)1023;
    float* dis  = ws;
    float* bufA = ws + disPad;
    float* bufB = bufA + ND;
    float* bufC = bufB + ND;

    // Zero accumulators (graph-capturable async memsets).
    hipMemsetAsync(dis,  0, (size_t)N * sizeof(float), stream);
    hipMemsetAsync(bufB, 0, ND * sizeof(float), stream);
    hipMemsetAsync(bufC, 0, ND * sizeof(float), stream);

    const int n4 = (int)(ND / 4);
    const int gElem  = (n4 + 255) / 256;
    const int gEdgeW = (int)(((size_t)E * 32 + 255) / 256);
    const int gRowT  = (N + 15) / 16;

    // 1) degree + normalization
    degree_kernel<<<(E + 255) / 256, 256, 0, stream>>>(col, ew, dis, E);
    dis_kernel<<<(N + 255) / 256, 256, 0, stream>>>(dis, N);

    // 2) h = emb @ W0
    gemm_n128_kernel<<<gRowT, 256, 0, stream>>>(emb, W0, bufA, N);

    // 3) agg1 = scatter-add(norm * h[row] -> col)
    aggregate_kernel<<<gEdgeW, 256, 0, stream>>>(row, col, ew, dis, bufA, bufB, E);

    // 4) h1 = elu(BN(agg1 + dis^2*h + b0))   (self-loop fused)
    epilogue1_kernel<<<gElem, 256, 0, stream>>>(bufB, bufA, dis, b0, bng, bnb, bnm, bnv, n4);

    // 5) h2pre = h1 @ W1
    gemm_n128_kernel<<<gRowT, 256, 0, stream>>>(bufB, W1, bufA, N);

    // 6) agg2 = scatter-add(norm * h2pre[row] -> col)
    aggregate_kernel<<<gEdgeW, 256, 0, stream>>>(row, col, ew, dis, bufA, bufC, E);

    // 7) out = (emb + h1 + (agg2 + dis^2*h2pre + b1)) / 3
    final_kernel<<<gElem, 256, 0, stream>>>(emb, bufB, bufC, bufA, dis, b1, out, n4);
}